// PrimitiveFitting_3058016715156
// MI455X (gfx1250) — compile-verified
//
#include <hip/hip_runtime.h>
#include <hip/hip_bf16.h>
#include <math.h>

typedef __attribute__((ext_vector_type(2))) float v2f;
typedef __attribute__((ext_vector_type(8))) float v8f;

#define GRIDSZ 0.4f

// ---------------- helpers ----------------
__device__ __forceinline__ unsigned encF(float f) {
    unsigned u = __float_as_uint(f);
    return (u & 0x80000000u) ? ~u : (u | 0x80000000u);   // order-preserving float->uint
}
__device__ __forceinline__ float decF(unsigned e) {
    unsigned b = (e & 0x80000000u) ? (e & 0x7FFFFFFFu) : ~e;
    return __uint_as_float(b);
}

// ---- workspace header (u32 units) ----
// hdrU[0..3]  encoded component mins (b,x,y,z)
// hdrU[4..7]  encoded component maxs
// hdrI[16..19] dims[0..3]
// hdrI[20] sx = dims2*dims3 ; hdrI[21] sy = dims3
// hdrF[32..35] pc_min[0..3] ; hdrF[36] decay2 ; hdrF[37] = -1/decay2

__global__ void k_init_hdr(unsigned* hdrU) {
    int t = threadIdx.x;
    if (t < 4) hdrU[t] = 0xFFFFFFFFu;
    else if (t < 8) hdrU[t] = 0u;
}

__global__ void k_zero(float* p, int total) {
    int i = blockIdx.x * blockDim.x + threadIdx.x;
    if (i < total) p[i] = 0.f;
}

__global__ void k_minmax(const float4* __restrict__ pts, int n, unsigned* hdrU) {
    int i0 = blockIdx.x * blockDim.x + threadIdx.x;
    int stride = gridDim.x * blockDim.x;
    float mn[4] = {1e30f, 1e30f, 1e30f, 1e30f};
    float mx[4] = {-1e30f, -1e30f, -1e30f, -1e30f};
    for (int i = i0; i < n; i += stride) {
        float4 p = pts[i];
        float c[4] = {p.x, p.y, p.z, p.w};
#pragma unroll
        for (int k = 0; k < 4; ++k) { mn[k] = fminf(mn[k], c[k]); mx[k] = fmaxf(mx[k], c[k]); }
    }
    // wave32 reduction
#pragma unroll
    for (int off = 16; off > 0; off >>= 1) {
#pragma unroll
        for (int k = 0; k < 4; ++k) {
            mn[k] = fminf(mn[k], __shfl_down(mn[k], off, 32));
            mx[k] = fmaxf(mx[k], __shfl_down(mx[k], off, 32));
        }
    }
    if ((threadIdx.x & 31) == 0) {
#pragma unroll
        for (int k = 0; k < 4; ++k) {
            atomicMin(&hdrU[k],     encF(mn[k]));
            atomicMax(&hdrU[4 + k], encF(mx[k]));
        }
    }
}

__global__ void k_setup(unsigned* hdrU) {
    if (threadIdx.x != 0 || blockIdx.x != 0) return;
    int*   hdrI = (int*)hdrU;
    float* hdrF = (float*)hdrU;          // only slots >= 32 accessed as float
    float vs[4] = {1.f, GRIDSZ, GRIDSZ, GRIDSZ};
    int dims[4];
#pragma unroll
    for (int k = 0; k < 4; ++k) {
        float mn = decF(hdrU[k])     - 2.f * vs[k];
        float mx = decF(hdrU[4 + k]) + 2.f * vs[k];
        dims[k] = (int)rintf((mx - mn) / vs[k]) + 3;
        hdrF[32 + k] = mn;
        hdrI[16 + k] = dims[k];
    }
    hdrI[20] = dims[2] * dims[3];
    hdrI[21] = dims[3];
    float decay2 = (3.f * GRIDSZ * GRIDSZ) * 0.25f;      // decay_radius^2
    hdrF[36] = decay2;
    hdrF[37] = -1.f / decay2;                            // hoist divide out of inner loop
}

__global__ void k_keys(const float4* __restrict__ pts, int n, int npow2,
                       const unsigned* __restrict__ hdrU, int* __restrict__ keys) {
    int i = blockIdx.x * blockDim.x + threadIdx.x;
    if (i >= npow2) return;
    if (i >= n) { keys[i] = 0x7FFFFFFF; return; }       // sort sentinel
    const int*   hdrI = (const int*)hdrU;
    const float* hdrF = (const float*)hdrU;
    float4 p = pts[i];                                   // (b,x,y,z)
    int c0 = (int)rintf( p.x - hdrF[32]) + 1;            // vs0 = 1
    int c1 = (int)rintf((p.y - hdrF[33]) / GRIDSZ) + 1;
    int c2 = (int)rintf((p.z - hdrF[34]) / GRIDSZ) + 1;
    int c3 = (int)rintf((p.w - hdrF[35]) / GRIDSZ) + 1;
    keys[i] = ((c0 * hdrI[17] + c1) * hdrI[18] + c2) * hdrI[19] + c3;
}

__global__ void k_bitonic(int* __restrict__ a, int j, int k) {
    unsigned i = blockIdx.x * blockDim.x + threadIdx.x;
    unsigned ixj = i ^ (unsigned)j;
    if (ixj > i) {
        int x = a[i], y = a[ixj];
        bool up = ((i & (unsigned)k) == 0);
        if ((x > y) == up) { a[i] = y; a[ixj] = x; }
    }
}

__global__ void k_fill_uq(int* __restrict__ uq, const int* __restrict__ keys, int n) {
    int i = blockIdx.x * blockDim.x + threadIdx.x;
    if (i < n) uq[i] = keys[n - 1];                      // fill = keys.max()
}

// per-block (1024 elems) inclusive rank of unique-flags + block totals
__global__ void k_scan_block(const int* __restrict__ keys, int* __restrict__ pos,
                             int* __restrict__ bsums, int n) {
    __shared__ int sh[256];
    int t = threadIdx.x;
    int base = blockIdx.x * 1024 + t * 4;
    int f[4], s = 0;
#pragma unroll
    for (int k = 0; k < 4; ++k) {
        int i = base + k;
        int fl = 0;
        if (i < n) fl = (i == 0 || keys[i] != keys[i - 1]) ? 1 : 0;
        f[k] = fl; s += fl;
    }
    sh[t] = s; __syncthreads();
    for (int off = 1; off < 256; off <<= 1) {
        int v = (t >= off) ? sh[t - off] : 0;
        __syncthreads();
        sh[t] += v;
        __syncthreads();
    }
    int run = sh[t] - s;                                 // exclusive prefix of this thread
#pragma unroll
    for (int k = 0; k < 4; ++k) {
        run += f[k];
        if (base + k < n) pos[base + k] = run;           // inclusive rank within block
    }
    if (t == 255) bsums[blockIdx.x] = sh[255];
}

__global__ void k_scan_sums(int* bsums, int nb) {
    if (threadIdx.x == 0 && blockIdx.x == 0) {
        int acc = 0;
        for (int i = 0; i < nb; ++i) { int v = bsums[i]; bsums[i] = acc; acc += v; }
    }
}

__global__ void k_uq_build(const int* __restrict__ keys, const int* __restrict__ pos,
                           const int* __restrict__ bsums, int* __restrict__ uq, int n) {
    int i = blockIdx.x * blockDim.x + threadIdx.x;
    if (i >= n) return;
    int ki = keys[i];
    if (i == 0 || keys[i - 1] != ki)
        uq[bsums[i >> 10] + pos[i] - 1] = ki;            // sorted unique, compacted
}

// ------------- scatter: 16 points/wave, neighbor cross-terms via V_WMMA_F32_16X16X4_F32 -------------
// d2(M,N) = ||dp_M||^2 - 2*(dp_M . o_N) + ||o_N||^2 ; cross term is a 16x27x3 GEMM done as
// two 16x16x4 WMMAs. A layout per ISA 7.12.2 (32-bit A 16x4): lanes 0-15 hold K0/K1, lanes
// 16-31 hold K2/K3 for the same M=lane%16. B mirrors (rows K0/K2 in vgpr0, K1/K3 in vgpr1).
// D layout: vgpr v -> M = v + 8*(lane/16), N = lane%16.
// Invalid tail points carry sentinel key 0x40000000 (>> any real key) so every neighbor
// lookup misses -- avoids shuffling a validity flag.
__global__ void k_scatter(const float4* __restrict__ pts, const int* __restrict__ uq,
                          float* __restrict__ wsum, float* __restrict__ wp,
                          float* __restrict__ wo6, const unsigned* __restrict__ hdrU,
                          int n, int nchunks) {
    int lane = threadIdx.x & 31;
    int wid  = (blockIdx.x * blockDim.x + threadIdx.x) >> 5;
    if (wid >= nchunks) return;                          // wave-uniform
    int half = lane >> 4, m = lane & 15;

    const int*   hdrI = (const int*)hdrU;
    const float* hdrF = (const float*)hdrU;
    float pminb = hdrF[32], pminx = hdrF[33], pminy = hdrF[34], pminz = hdrF[35];
    float nInvDecay2 = hdrF[37];
    int d1 = hdrI[17], d2d = hdrI[18], d3 = hdrI[19];
    int sx = hdrI[20], sy = hdrI[21];

    int pi = wid * 16 + m;
    bool pvalid = pi < n;
    float4 P = pvalid ? pts[pi] : make_float4(0.f, 0.f, 0.f, 0.f);
    float pb = P.x, px = P.y, py = P.z, pz = P.w;

    int c0 = (int)rintf( pb - pminb) + 1;
    int c1 = (int)rintf((px - pminx) / GRIDSZ) + 1;
    int c2 = (int)rintf((py - pminy) / GRIDSZ) + 1;
    int c3 = (int)rintf((pz - pminz) / GRIDSZ) + 1;
    int key = pvalid ? (((c0 * d1 + c1) * d2d + c2) * d3 + c3) : 0x40000000;

    float dpx = px - ((float)(c1 - 1) * GRIDSZ + pminx);
    float dpy = py - ((float)(c2 - 1) * GRIDSZ + pminy);
    float dpz = pz - ((float)(c3 - 1) * GRIDSZ + pminz);
    float dpn2 = dpx * dpx + dpy * dpy + dpz * dpz;

    // A fragment (16x4 f32): K0/K1 = dp.x/dp.y (lanes 0-15), K2/K3 = dp.z/0 (lanes 16-31)
    v2f afrag;
    afrag.x = half ? dpz : dpx;
    afrag.y = half ? 0.f : dpy;

    for (int g = 0; g < 2; ++g) {
        int j = g * 16 + m;                              // this lane's neighbor column
        bool jok = j < 27;
        int ox = jok ? (j / 9) - 1 : 0;
        int oy = jok ? ((j / 3) % 3) - 1 : 0;
        int oz = jok ? (j % 3) - 1 : 0;
        float Ox = (float)ox * GRIDSZ, Oy = (float)oy * GRIDSZ, Oz = (float)oz * GRIDSZ;

        v2f bfrag;                                       // B (4x16): rows K0/K2, K1/K3
        bfrag.x = half ? Oz : Ox;
        bfrag.y = half ? 0.f : Oy;

        v8f acc = {};
        acc = __builtin_amdgcn_wmma_f32_16x16x4_f32(false, afrag, false, bfrag,
                                                    (short)0, acc, false, false);
        union { v8f v; float f[8]; } au; au.v = acc;

        float on2 = Ox * Ox + Oy * Oy + Oz * Oz;
        int noff = ox * sx + oy * sy + oz;               // key delta of this neighbor

        for (int v = 0; v < 8; ++v) {
            int M = v + 8 * half;                        // point row of this D vgpr
            // all 32 lanes must execute the permutes (sources must be EXEC-enabled)
            int   kM = __shfl(key,  M, 32);
            float dn = __shfl(dpn2, M, 32);
            float vb = __shfl(pb,   M, 32);
            float vx = __shfl(px,   M, 32);
            float vy = __shfl(py,   M, 32);
            float vz = __shfl(pz,   M, 32);
            if (!jok) continue;

            int nkey = kM + noff;
            int lo = 0, hi = n;                          // lower_bound == searchsorted('left')
            while (lo < hi) { int mid = (lo + hi) >> 1; if (uq[mid] < nkey) lo = mid + 1; else hi = mid; }
            int idx = lo < n ? lo : n - 1;
            if (uq[idx] != nkey) continue;               // query voxel not occupied

            float dd2 = dn - 2.f * au.f[v] + on2;
            float w = __expf(dd2 * nInvDecay2);          // exp(-d2/decay2), no divide
            atomicAdd(&wsum[idx], w);
            atomicAdd(&wp[idx * 4 + 0], w * vb);
            atomicAdd(&wp[idx * 4 + 1], w * vx);
            atomicAdd(&wp[idx * 4 + 2], w * vy);
            atomicAdd(&wp[idx * 4 + 3], w * vz);
            atomicAdd(&wo6[idx * 6 + 0], w * vx * vx);
            atomicAdd(&wo6[idx * 6 + 1], w * vx * vy);
            atomicAdd(&wo6[idx * 6 + 2], w * vx * vz);
            atomicAdd(&wo6[idx * 6 + 3], w * vy * vy);
            atomicAdd(&wo6[idx * 6 + 4], w * vy * vz);
            atomicAdd(&wo6[idx * 6 + 5], w * vz * vz);
        }
    }
}

// ------------- finalize: weighted mean, covariance, 3x3 symmetric eigen (Jacobi) -------------
__global__ void k_finalize(const float* __restrict__ wsum, const float* __restrict__ wp,
                           const float* __restrict__ wo6, float* __restrict__ out, int n) {
    int j = blockIdx.x * blockDim.x + threadIdx.x;
    if (j >= n) return;
    float* mu = out + j * 4;
    float* R  = out + n * 4 + j * 9;
    float ws = wsum[j];
    if (!(ws > 0.f)) {                                   // !ok -> mu*0, R*0
#pragma unroll
        for (int k = 0; k < 4; ++k) mu[k] = 0.f;
#pragma unroll
        for (int k = 0; k < 9; ++k) R[k] = 0.f;
        return;
    }
    float inv = 1.f / ws;
    float m0 = wp[j * 4 + 0] * inv;
    float mx = wp[j * 4 + 1] * inv;
    float my = wp[j * 4 + 2] * inv;
    float mz = wp[j * 4 + 3] * inv;

    float a[3][3];
    a[0][0] = wo6[j * 6 + 0] * inv - mx * mx;
    a[0][1] = wo6[j * 6 + 1] * inv - mx * my;
    a[0][2] = wo6[j * 6 + 2] * inv - mx * mz;
    a[1][1] = wo6[j * 6 + 3] * inv - my * my;
    a[1][2] = wo6[j * 6 + 4] * inv - my * mz;
    a[2][2] = wo6[j * 6 + 5] * inv - mz * mz;
    a[1][0] = a[0][1]; a[2][0] = a[0][2]; a[2][1] = a[1][2];

    float V[3][3] = {{1.f, 0.f, 0.f}, {0.f, 1.f, 0.f}, {0.f, 0.f, 1.f}};
    const int PP[3] = {0, 0, 1}, QQ[3] = {1, 2, 2};
    for (int sweep = 0; sweep < 12; ++sweep) {
#pragma unroll
        for (int r = 0; r < 3; ++r) {
            int p = PP[r], q = QQ[r];
            float apq = a[p][q];
            if (fabsf(apq) > 1e-20f) {
                float theta = (a[q][q] - a[p][p]) / (2.f * apq);
                float t = copysignf(1.f, theta) / (fabsf(theta) + sqrtf(theta * theta + 1.f));
                float c = 1.f / sqrtf(t * t + 1.f);
                float s = t * c;
#pragma unroll
                for (int k = 0; k < 3; ++k) {
                    float akp = a[k][p], akq = a[k][q];
                    a[k][p] = c * akp - s * akq;
                    a[k][q] = s * akp + c * akq;
                }
#pragma unroll
                for (int k = 0; k < 3; ++k) {
                    float apk = a[p][k], aqk = a[q][k];
                    a[p][k] = c * apk - s * aqk;
                    a[q][k] = s * apk + c * aqk;
                }
#pragma unroll
                for (int k = 0; k < 3; ++k) {
                    float vkp = V[k][p], vkq = V[k][q];
                    V[k][p] = c * vkp - s * vkq;
                    V[k][q] = s * vkp + c * vkq;
                }
            }
        }
    }
    // singular values = |eigenvalues|, sorted descending, U columns follow
    float l0 = fabsf(a[0][0]), l1 = fabsf(a[1][1]), l2 = fabsf(a[2][2]);
    int i0 = 0, i1 = 1, i2 = 2;
    if (l0 < l1) { float t = l0; l0 = l1; l1 = t; int ti = i0; i0 = i1; i1 = ti; }
    if (l1 < l2) { float t = l1; l1 = l2; l2 = t; int ti = i1; i1 = i2; i2 = ti; }
    if (l0 < l1) { float t = l0; l0 = l1; l1 = t; int ti = i0; i0 = i1; i1 = ti; }
    float U[3][3];
#pragma unroll
    for (int r = 0; r < 3; ++r) { U[r][0] = V[r][i0]; U[r][1] = V[r][i1]; U[r][2] = V[r][i2]; }
    float det = U[0][0] * (U[1][1] * U[2][2] - U[1][2] * U[2][1])
              - U[0][1] * (U[1][0] * U[2][2] - U[1][2] * U[2][0])
              + U[0][2] * (U[1][0] * U[2][1] - U[1][1] * U[2][0]);
    float flip = (det < 0.f) ? -1.f : 1.f;
    float S0 = fmaxf(sqrtf(l0), 1e-6f);
    float S1 = fmaxf(sqrtf(l1), 1e-6f);
    float S2 = fmaxf(sqrtf(l2), 1e-6f) * flip;

    mu[0] = m0; mu[1] = mx; mu[2] = my; mu[3] = mz;
#pragma unroll
    for (int r = 0; r < 3; ++r) {
        R[r * 3 + 0] = U[r][0] * S0;
        R[r * 3 + 1] = U[r][1] * S1;
        R[r * 3 + 2] = U[r][2] * S2;
    }
}

// ---------------- host launcher ----------------
extern "C" void kernel_launch(void* const* d_in, const int* in_sizes, int n_in,
                              void* d_out, int out_size, void* d_ws, size_t ws_size,
                              hipStream_t stream) {
    const float4* pts = (const float4*)d_in[0];
    float* out = (float*)d_out;
    int n = in_sizes[0] / 4;
    int npow2 = 1; while (npow2 < n) npow2 <<= 1;
    int nb_scan = (n + 1023) / 1024;

    char* w = (char*)d_ws;
    unsigned* hdrU  = (unsigned*)w;                      // 64 u32 slots = 256 B
    int*      keys  = (int*)(w + 256);
    int*      pos   = keys + npow2;
    int*      bsums = pos + n;
    int*      uq    = bsums + nb_scan;
    float*    wsumA = (float*)(uq + n);
    float*    wpA   = wsumA + n;                         // n*4
    float*    wo6A  = wpA + 4 * n;                       // n*6  (contiguous block of 11*n floats)

    k_init_hdr<<<1, 64, 0, stream>>>(hdrU);
    k_zero<<<(11 * n + 255) / 256, 256, 0, stream>>>(wsumA, 11 * n);
    k_minmax<<<256, 256, 0, stream>>>(pts, n, hdrU);
    k_setup<<<1, 1, 0, stream>>>(hdrU);
    k_keys<<<(npow2 + 255) / 256, 256, 0, stream>>>(pts, n, npow2, hdrU, keys);

    for (int k = 2; k <= npow2; k <<= 1)
        for (int j = k >> 1; j > 0; j >>= 1)
            k_bitonic<<<npow2 / 256, 256, 0, stream>>>(keys, j, k);

    k_fill_uq<<<(n + 255) / 256, 256, 0, stream>>>(uq, keys, n);
    k_scan_block<<<nb_scan, 256, 0, stream>>>(keys, pos, bsums, n);
    k_scan_sums<<<1, 1, 0, stream>>>(bsums, nb_scan);
    k_uq_build<<<(n + 255) / 256, 256, 0, stream>>>(keys, pos, bsums, uq, n);

    int chunks = (n + 15) / 16;                          // 16 points per wave
    int sblocks = (chunks * 32 + 255) / 256;
    k_scatter<<<sblocks, 256, 0, stream>>>(pts, uq, wsumA, wpA, wo6A, hdrU, n, chunks);

    k_finalize<<<(n + 255) / 256, 256, 0, stream>>>(wsumA, wpA, wo6A, out, n);
}